// GNN_v4_5927054868947
// MI455X (gfx1250) — compile-verified
//
#include <hip/hip_runtime.h>
#include <math.h>

#define NN 16384            // nodes
#define NE 65536            // edges
#define NG 64               // graphs
#define ESL (NE + NN)       // edges + self loops (GATv2)

typedef float v2f __attribute__((ext_vector_type(2)));
typedef float v8f __attribute__((ext_vector_type(8)));

// ---------------------------------------------------------------- utilities
__device__ __forceinline__ void atomicMaxF(float* addr, float val) {
  int* ai = reinterpret_cast<int*>(addr);
  int old = __float_as_int(*addr);
  while (__int_as_float(old) < val) {
    int assumed = old;
    old = atomicCAS(ai, assumed, __float_as_int(val));
    if (old == assumed) break;
  }
}

__global__ void fill_f(float* p, float v, size_t n) {
  size_t i = (size_t)blockIdx.x * blockDim.x + threadIdx.x;
  if (i < n) p[i] = v;
}

// ---------------------------------------------------------------- WMMA GEMM
// C[M,N] = H[M,K] @ W[K,N]  (row-major fp32; K%32==0, N%64==0, M%128==0)
// Block = 256 threads = 8 waves; BM=128, BN=64, BK=32.
// A (128x32) and B (32x64) tiles staged in LDS (B shared by all 8 waves),
// each wave computes a 16x64 strip via 4 x V_WMMA_F32_16X16X4_F32 per K-step.
#define BM 128
#define BN 64
#define BK 32
#define AS_STR 34   // pad: 34*lm mod 64 -> 16 distinct banks; even -> b64-aligned pairs
#define BS_STR 66

__global__ __launch_bounds__(256)
void gemm_wmma(const float* __restrict__ H, const float* __restrict__ W,
               float* __restrict__ C, int M, int K, int N) {
  __shared__ float As[BM * AS_STR];
  __shared__ float Bs[BK * BS_STR];
  const int lane = threadIdx.x;                 // 0..31
  const int wv   = threadIdx.y;                 // 0..7
  const int tid  = wv * 32 + lane;              // 0..255
  const int mblk = blockIdx.x * BM;
  const int n0   = blockIdx.y * BN;
  const int lm    = lane & 15;
  const int khalf = (lane >> 4) << 1;           // 0 or 2

  v8f c0 = {}, c1 = {}, c2 = {}, c3 = {};
  for (int k0 = 0; k0 < K; k0 += BK) {
    // ---- stage A tile: 128x32 floats = 1024 float4, 4 per thread (coalesced)
#pragma unroll
    for (int i = 0; i < 4; ++i) {
      int q = tid + i * 256;                    // float4 chunk id
      int r = q >> 3, c = (q & 7) << 2;
      const float* g = H + (size_t)(mblk + r) * K + k0 + c;
      float4 v = *(const float4*)g;
      if (i == 0 && k0 + BK < K) __builtin_prefetch(g + BK, 0, 3);
      float* sp = As + r * AS_STR + c;
      sp[0] = v.x; sp[1] = v.y; sp[2] = v.z; sp[3] = v.w;
    }
    // ---- stage B tile: 32x64 floats = 512 float4, 2 per thread (coalesced)
#pragma unroll
    for (int i = 0; i < 2; ++i) {
      int q = tid + i * 256;
      int r = q >> 4, c = (q & 15) << 2;
      const float* g = W + (size_t)(k0 + r) * N + n0 + c;
      float4 v = *(const float4*)g;
      if (i == 0 && k0 + BK < K) __builtin_prefetch(g + (size_t)BK * N, 0, 3);
      float* sp = Bs + r * BS_STR + c;
      sp[0] = v.x; sp[1] = v.y; sp[2] = v.z; sp[3] = v.w;
    }
    __syncthreads();
    const float* arow = As + (wv * 16 + lm) * AS_STR;
#pragma unroll
    for (int kk = 0; kk < BK; kk += 4) {
      v2f a;
      a.x = arow[kk + khalf];
      a.y = arow[kk + khalf + 1];
      const float* b0p = Bs + (kk + khalf) * BS_STR + lm;
      const float* b1p = b0p + BS_STR;
      v2f b0, b1, b2, b3;
      b0.x = b0p[0];  b0.y = b1p[0];
      b1.x = b0p[16]; b1.y = b1p[16];
      b2.x = b0p[32]; b2.y = b1p[32];
      b3.x = b0p[48]; b3.y = b1p[48];
      c0 = __builtin_amdgcn_wmma_f32_16x16x4_f32(false, a, false, b0, (short)0, c0, false, false);
      c1 = __builtin_amdgcn_wmma_f32_16x16x4_f32(false, a, false, b1, (short)0, c1, false, false);
      c2 = __builtin_amdgcn_wmma_f32_16x16x4_f32(false, a, false, b2, (short)0, c2, false, false);
      c3 = __builtin_amdgcn_wmma_f32_16x16x4_f32(false, a, false, b3, (short)0, c3, false, false);
    }
    __syncthreads();
  }
  // ---- store: VGPR r holds C[M = r + 8*(lane>=16)][N = lane%16 (+0/16/32/48)]
  const int ms = mblk + wv * 16 + ((lane >> 4) << 3);
  float* crow = C + (size_t)ms * N + n0 + lm;
#pragma unroll
  for (int r = 0; r < 8; ++r) {
    crow[(size_t)r * N]      = c0[r];
    crow[(size_t)r * N + 16] = c1[r];
    crow[(size_t)r * N + 32] = c2[r];
    crow[(size_t)r * N + 48] = c3[r];
  }
}

static inline void launch_gemm(const float* H, const float* W, float* C,
                               int M, int K, int N, hipStream_t s) {
  dim3 blk(32, 8);
  dim3 grd(M / BM, N / BN);
  gemm_wmma<<<grd, blk, 0, s>>>(H, W, C, M, K, N);
}

// ---------------------------------------------------------------- graph prep
__global__ void count_batch(const int* batch, float* cnt) {
  int i = blockIdx.x * blockDim.x + threadIdx.x;
  if (i < NN) atomicAdd(&cnt[batch[i]], 1.0f);
}
__global__ void clamp_cnt(float* cnt) {
  int i = threadIdx.x;
  if (i < NG) cnt[i] = fmaxf(cnt[i], 1.0f);
}
__global__ void count_deg(const int* dst, float* deg) {
  int e = blockIdx.x * blockDim.x + threadIdx.x;
  if (e < NE) atomicAdd(&deg[dst[e]], 1.0f);
}
__global__ void make_dis(float* deg) {  // in-place deg -> 1/sqrt(deg) (0 if deg==0)
  int i = blockIdx.x * blockDim.x + threadIdx.x;
  if (i < NN) {
    float d = deg[i];
    deg[i] = d > 0.0f ? 1.0f / sqrtf(fmaxf(d, 1.0f)) : 0.0f;
  }
}
__global__ void edge_norm(const int* src, const int* dst, const float* dis, float* nrm) {
  int e = blockIdx.x * blockDim.x + threadIdx.x;
  if (e < NE) nrm[e] = dis[src[e]] * dis[dst[e]];
}

// ---------------------------------------------------------------- GAT pieces
__global__ void outer_1(const float* x, const float* wl, const float* wr,
                        float* xl, float* xr) {  // GAT1: K=1 outer product
  int i = blockIdx.x * blockDim.x + threadIdx.x;
  if (i >= NN * 64) return;
  int n = i >> 6, j = i & 63;
  float v = x[n];
  xl[i] = v * wl[j];
  xr[i] = v * wr[j];
}

// e = att . leaky_relu(xl[src] + xr[dst]);  one wave per edge, lanes stride channels
__global__ void gat_score(const float* __restrict__ xl, const float* __restrict__ xr,
                          const float* __restrict__ att, const int* src, const int* dst,
                          int Fo, float* ebuf, float* nmax) {
  int lane = threadIdx.x & 31, wv = threadIdx.x >> 5;
  int e = blockIdx.x * 8 + wv;
  if (e >= ESL) return;
  int s, d;
  if (e < NE) { s = src[e]; d = dst[e]; } else { s = d = e - NE; }
  const float* pl = xl + (size_t)s * Fo;
  const float* pr = xr + (size_t)d * Fo;
  float acc = 0.0f;
  for (int j = lane; j < Fo; j += 32) {
    float v = pl[j] + pr[j];
    v = v > 0.0f ? v : 0.2f * v;
    acc += att[j] * v;
  }
  for (int o = 16; o > 0; o >>= 1) acc += __shfl_down(acc, o, 32);
  if (lane == 0) { ebuf[e] = acc; atomicMaxF(&nmax[d], acc); }
}

__global__ void gat_expsum(const float* ebuf, const int* dst, const float* nmax,
                           float* abuf, float* nsum) {
  int e = blockIdx.x * blockDim.x + threadIdx.x;
  if (e >= ESL) return;
  int d = e < NE ? dst[e] : e - NE;
  float ex = expf(ebuf[e] - nmax[d]);
  abuf[e] = ex;
  atomicAdd(&nsum[d], ex);
}

__global__ void gat_scatter(const float* __restrict__ xl, const float* abuf,
                            const float* nsum, const int* src, const int* dst,
                            int Fo, float* __restrict__ out) {
  int lane = threadIdx.x & 31, wv = threadIdx.x >> 5;
  int e = blockIdx.x * 8 + wv;
  if (e >= ESL) return;
  int s, d;
  if (e < NE) { s = src[e]; d = dst[e]; } else { s = d = e - NE; }
  float a = abuf[e] / (nsum[d] + 1e-16f);
  const float* pl = xl + (size_t)s * Fo;
  float* po = out + (size_t)d * Fo;
  for (int j = lane; j < Fo; j += 32) atomicAdd(&po[j], pl[j] * a);
}

__global__ void bias_elu(float* X, const float* b, int C, size_t total) {
  size_t i = (size_t)blockIdx.x * blockDim.x + threadIdx.x;
  if (i >= total) return;
  float v = X[i] + b[i % C];
  X[i] = v > 0.0f ? v : (expf(v) - 1.0f);
}

// ---------------------------------------------------------------- GraphNorm
__global__ void gn_mean(const float* X, const int* batch, float* gmean, int C, size_t total) {
  size_t i = (size_t)blockIdx.x * blockDim.x + threadIdx.x;
  if (i >= total) return;
  int node = (int)(i / C), j = (int)(i % C);
  atomicAdd(&gmean[(size_t)batch[node] * C + j], X[i]);
}
__global__ void gn_center(float* X, const int* batch, const float* gmean,
                          const float* cnt, const float* ms, float* gvar,
                          int C, size_t total) {
  size_t i = (size_t)blockIdx.x * blockDim.x + threadIdx.x;
  if (i >= total) return;
  int node = (int)(i / C), j = (int)(i % C);
  int g = batch[node];
  float m = gmean[(size_t)g * C + j] / cnt[g];
  float t = X[i] - m * ms[j];
  X[i] = t;
  atomicAdd(&gvar[(size_t)g * C + j], t * t);
}
__global__ void gn_final(float* X, const int* batch, const float* gvar,
                         const float* cnt, const float* w, const float* b,
                         int C, size_t total) {
  size_t i = (size_t)blockIdx.x * blockDim.x + threadIdx.x;
  if (i >= total) return;
  int node = (int)(i / C), j = (int)(i % C);
  int g = batch[node];
  float var = gvar[(size_t)g * C + j] / cnt[g];
  X[i] = w[j] * X[i] / sqrtf(var + 1e-5f) + b[j];
}

// ---------------------------------------------------------------- ARMA pieces
__global__ void arma_scatter(const float* __restrict__ t, const float* nrm,
                             const int* src, const int* dst, int Fo,
                             float* __restrict__ agg) {
  int lane = threadIdx.x & 31, wv = threadIdx.x >> 5;
  int e = blockIdx.x * 8 + wv;
  if (e >= NE) return;
  int s = src[e], d = dst[e];
  float nm = nrm[e];
  const float* ts = t + (size_t)s * Fo;
  float* ad = agg + (size_t)d * Fo;
  for (int j = lane; j < Fo; j += 32) atomicAdd(&ad[j], ts[j] * nm);
}
__global__ void arma_combine(float* S, const float* u, const float* b, int C, size_t total) {
  size_t i = (size_t)blockIdx.x * blockDim.x + threadIdx.x;
  if (i >= total) return;
  float v = S[i] + u[i] + b[i % C];
  S[i] = v > 0.0f ? v : 0.0f;   // relu; outer elu is identity on >=0
}

// ---------------------------------------------------------------- readout
__global__ void pool_kernel(const float* X, const int* batch, float* gmax, float* gsum) {
  int i = blockIdx.x * blockDim.x + threadIdx.x;
  if (i >= NN * 64) return;
  int n = i >> 6, j = i & 63;
  int g = batch[n];
  float v = X[i];
  atomicMaxF(&gmax[g * 64 + j], v);
  atomicAdd(&gsum[g * 64 + j], v);
}
__global__ void final_linear(const float* gmax, const float* gsum, const float* cnt,
                             const float* lw, const float* lb, float* out) {
  int t = threadIdx.x;
  if (t >= NG * 2) return;
  int g = t >> 1, o = t & 1;
  float c = cnt[g];
  float acc = lb[o];
  for (int j = 0; j < 64; ++j) {
    float mx = gmax[g * 64 + j];
    float sm = gsum[g * 64 + j];
    acc += mx * lw[j * 2 + o] + (sm / c) * lw[(64 + j) * 2 + o] + sm * lw[(128 + j) * 2 + o];
  }
  out[t] = acc;
}

// ---------------------------------------------------------------- host side
static inline int nb(size_t n) { return (int)((n + 255) / 256); }

struct Scratch {
  float *A, *B, *S1, *S2, *ebuf, *abuf, *nrm, *nmax, *nsum, *dis, *cnt;
  float *gmean, *gvar, *gmax, *gsum;
};

static void run_gat(const float* hin, int Fi, int Fo,
                    const float* wl, const float* wr, const float* att, const float* bias,
                    const int* src, const int* dst, const int* batch,
                    float* xl, float* xr, float* out, const Scratch& S, hipStream_t s) {
  if (Fi == 1) {
    outer_1<<<nb((size_t)NN * 64), 256, 0, s>>>(hin, wl, wr, xl, xr);
  } else {
    launch_gemm(hin, wl, xl, NN, Fi, Fo, s);
    launch_gemm(hin, wr, xr, NN, Fi, Fo, s);
  }
  fill_f<<<nb(NN), 256, 0, s>>>(S.nmax, -INFINITY, NN);
  fill_f<<<nb(NN), 256, 0, s>>>(S.nsum, 0.0f, NN);
  gat_score<<<ESL / 8, 256, 0, s>>>(xl, xr, att, src, dst, Fo, S.ebuf, S.nmax);
  gat_expsum<<<nb(ESL), 256, 0, s>>>(S.ebuf, dst, S.nmax, S.abuf, S.nsum);
  size_t total = (size_t)NN * Fo;
  fill_f<<<nb(total), 256, 0, s>>>(out, 0.0f, total);     // out may alias xr: done after expsum
  gat_scatter<<<ESL / 8, 256, 0, s>>>(xl, S.abuf, S.nsum, src, dst, Fo, out);
  bias_elu<<<nb(total), 256, 0, s>>>(out, bias, Fo, total);
}

static void run_gn(float* X, int C, const int* batch, const float* w, const float* b,
                   const float* ms, const Scratch& S, hipStream_t s) {
  size_t gc = (size_t)NG * C, total = (size_t)NN * C;
  fill_f<<<nb(gc), 256, 0, s>>>(S.gmean, 0.0f, gc);
  fill_f<<<nb(gc), 256, 0, s>>>(S.gvar, 0.0f, gc);
  gn_mean<<<nb(total), 256, 0, s>>>(X, batch, S.gmean, C, total);
  gn_center<<<nb(total), 256, 0, s>>>(X, batch, S.gmean, S.cnt, ms, S.gvar, C, total);
  gn_final<<<nb(total), 256, 0, s>>>(X, batch, S.gvar, S.cnt, w, b, C, total);
}

extern "C" void kernel_launch(void* const* d_in, const int* in_sizes, int n_in,
                              void* d_out, int out_size, void* d_ws, size_t ws_size,
                              hipStream_t stream) {
  const float* x    = (const float*)d_in[0];
  const int* src    = (const int*)d_in[1];
  const int* dst    = src + NE;
  const int* batch  = (const int*)d_in[2];
  const float* gat_wl[3]  = {(const float*)d_in[3],  (const float*)d_in[7],  (const float*)d_in[11]};
  const float* gat_wr[3]  = {(const float*)d_in[4],  (const float*)d_in[8],  (const float*)d_in[12]};
  const float* gat_att[3] = {(const float*)d_in[5],  (const float*)d_in[9],  (const float*)d_in[13]};
  const float* gat_b[3]   = {(const float*)d_in[6],  (const float*)d_in[10], (const float*)d_in[14]};
  const float* gn_w[5]  = {(const float*)d_in[15], (const float*)d_in[18], (const float*)d_in[21],
                           (const float*)d_in[24], (const float*)d_in[27]};
  const float* gn_b[5]  = {(const float*)d_in[16], (const float*)d_in[19], (const float*)d_in[22],
                           (const float*)d_in[25], (const float*)d_in[28]};
  const float* gn_ms[5] = {(const float*)d_in[17], (const float*)d_in[20], (const float*)d_in[23],
                           (const float*)d_in[26], (const float*)d_in[29]};
  const float* arma_wi[2] = {(const float*)d_in[30], (const float*)d_in[33]};
  const float* arma_wr[2] = {(const float*)d_in[31], (const float*)d_in[34]};
  const float* arma_b[2]  = {(const float*)d_in[32], (const float*)d_in[35]};
  const float* lin_w = (const float*)d_in[36];
  const float* lin_b = (const float*)d_in[37];
  float* outp = (float*)d_out;

  // workspace carve (~308 MB of fp32)
  float* ws = (float*)d_ws;
  size_t off = 0;
  Scratch S;
  S.A    = ws + off; off += (size_t)NN * 2048;
  S.B    = ws + off; off += (size_t)NN * 2048;
  S.S1   = ws + off; off += (size_t)NN * 64;
  S.S2   = ws + off; off += (size_t)NN * 512;
  S.ebuf = ws + off; off += ESL;
  S.abuf = ws + off; off += ESL;
  S.nrm  = ws + off; off += NE;
  S.nmax = ws + off; off += NN;
  S.nsum = ws + off; off += NN;
  S.dis  = ws + off; off += NN;
  S.cnt  = ws + off; off += NG;
  S.gmean= ws + off; off += (size_t)NG * 2048;
  S.gvar = ws + off; off += (size_t)NG * 2048;
  S.gmax = ws + off; off += (size_t)NG * 64;
  S.gsum = ws + off; off += (size_t)NG * 64;

  // ---- graph-level constants: cnt, gcn norm ----
  fill_f<<<1, 64, 0, stream>>>(S.cnt, 0.0f, NG);
  fill_f<<<nb(NN), 256, 0, stream>>>(S.dis, 0.0f, NN);
  count_batch<<<nb(NN), 256, 0, stream>>>(batch, S.cnt);
  count_deg<<<nb(NE), 256, 0, stream>>>(dst, S.dis);
  clamp_cnt<<<1, 64, 0, stream>>>(S.cnt);
  make_dis<<<nb(NN), 256, 0, stream>>>(S.dis);
  edge_norm<<<nb(NE), 256, 0, stream>>>(src, dst, S.dis, S.nrm);

  // ---- GAT1 (1 -> 64) -> S1 ----
  run_gat(x, 1, 64, gat_wl[0], gat_wr[0], gat_att[0], gat_b[0],
          src, dst, batch, S.A, S.B, S.S1, S, stream);
  run_gn(S.S1, 64, batch, gn_w[0], gn_b[0], gn_ms[0], S, stream);

  // ---- GAT2 (64 -> 512) -> S2 ----
  run_gat(S.S1, 64, 512, gat_wl[1], gat_wr[1], gat_att[1], gat_b[1],
          src, dst, batch, S.A, S.B, S.S2, S, stream);
  run_gn(S.S2, 512, batch, gn_w[1], gn_b[1], gn_ms[1], S, stream);

  // ---- GAT3 (512 -> 2048) -> B (out overwrites xr after softmax stats) ----
  run_gat(S.S2, 512, 2048, gat_wl[2], gat_wr[2], gat_att[2], gat_b[2],
          src, dst, batch, S.A, S.B, S.B, S, stream);
  run_gn(S.B, 2048, batch, gn_w[2], gn_b[2], gn_ms[2], S, stream);

  // ---- ARMA1 (2048 -> 512) -> S2 ----
  {
    float* t = S.A;                          // NN x 512
    float* u = S.A + (size_t)NN * 512;       // NN x 512
    launch_gemm(S.B, arma_wi[0], t, NN, 2048, 512, stream);
    launch_gemm(S.B, arma_wr[0], u, NN, 2048, 512, stream);
    size_t total = (size_t)NN * 512;
    fill_f<<<nb(total), 256, 0, stream>>>(S.S2, 0.0f, total);
    arma_scatter<<<NE / 8, 256, 0, stream>>>(t, S.nrm, src, dst, 512, S.S2);
    arma_combine<<<nb(total), 256, 0, stream>>>(S.S2, u, arma_b[0], 512, total);
    run_gn(S.S2, 512, batch, gn_w[3], gn_b[3], gn_ms[3], S, stream);
  }

  // ---- ARMA2 (512 -> 64) -> S1 ----
  {
    float* t = S.A;                          // NN x 64
    float* u = S.A + (size_t)NN * 64;        // NN x 64
    launch_gemm(S.S2, arma_wi[1], t, NN, 512, 64, stream);
    launch_gemm(S.S2, arma_wr[1], u, NN, 512, 64, stream);
    size_t total = (size_t)NN * 64;
    fill_f<<<nb(total), 256, 0, stream>>>(S.S1, 0.0f, total);
    arma_scatter<<<NE / 8, 256, 0, stream>>>(t, S.nrm, src, dst, 64, S.S1);
    arma_combine<<<nb(total), 256, 0, stream>>>(S.S1, u, arma_b[1], 64, total);
    run_gn(S.S1, 64, batch, gn_w[4], gn_b[4], gn_ms[4], S, stream);
  }

  // ---- readout: [max | mean | sum] @ lin_w + lin_b ----
  fill_f<<<nb((size_t)NG * 64), 256, 0, stream>>>(S.gmax, -INFINITY, (size_t)NG * 64);
  fill_f<<<nb((size_t)NG * 64), 256, 0, stream>>>(S.gsum, 0.0f, (size_t)NG * 64);
  pool_kernel<<<nb((size_t)NN * 64), 256, 0, stream>>>(S.S1, batch, S.gmax, S.gsum);
  final_linear<<<1, 128, 0, stream>>>(S.gmax, S.gsum, S.cnt, lin_w, lin_b, outp);
}